// LinearRecurrentUnitCell_12799002542673
// MI455X (gfx1250) — compile-verified
//
#include <hip/hip_runtime.h>
#include <stdint.h>

typedef __attribute__((ext_vector_type(2))) float        v2f;
typedef __attribute__((ext_vector_type(8))) float        v8f;
typedef __attribute__((ext_vector_type(4))) unsigned int u32x4;
typedef __attribute__((ext_vector_type(8))) int          i32x8;
typedef __attribute__((ext_vector_type(4))) int          i32x4;

#define BATCH 8192
#define NREC  1024
#define DHID  2048

#define TILE_K     32
#define BLOCK_ROWS 128              // 8 waves * 16 rows
#define LDS_AS     34               // padded LDS stride for A tiles (bank-conflict free)

#define S1_AS  (BLOCK_ROWS * LDS_AS)   // 4352 floats per A-tile buffer
#define S_BS   (TILE_K * 16)           // 512 floats per B/C-tile buffer

// Toolchain arity dispatch for the TDM builtin (see CDNA5_HIP.md):
//   amdgpu-toolchain (therock headers present) -> 6-arg form
//   ROCm 7.2                                   -> 5-arg form
#if defined(__has_include)
#  if __has_include(<hip/amd_detail/amd_gfx1250_TDM.h>)
#    define LRU_TDM_ARITY6 1
#  endif
#endif

__device__ __forceinline__ v8f wmma_f32(v2f a, v2f b, v8f c) {
    // V_WMMA_F32_16X16X4_F32:  D = A(16x4) * B(4x16) + C(16x16), all fp32
    return __builtin_amdgcn_wmma_f32_16x16x4_f32(
        /*neg_a=*/false, a, /*neg_b=*/false, b,
        /*c_mod=*/(short)0, c, /*reuse_a=*/false, /*reuse_b=*/false);
}

// The TDM builtin's LDS writes are not modeled as aliasing the __shared__
// array; without this the compiler deletes every ds_load of smem (it sees a
// never-stored LDS global). Passing the pointer into an empty asm with a
// memory clobber makes smem "escape" so its contents may change externally.
__device__ __forceinline__ void lds_publish(const float* p) {
    asm volatile("" : : "r"(p) : "memory");
}

// ---------------------------------------------------------------------------
// TDM: DMA a 2-D fp32 tile (tile_x cols x tile_y rows) from a row-major
// tensor in global memory into LDS. Optional padding inserts 2 dwords after
// every 32 dwords written (turning a 32-float tile row into the 34-float
// LDS stride the compute loop expects).
// D# packing per CDNA5 ISA.
// ---------------------------------------------------------------------------
__device__ __forceinline__ void tdm_load_tile(
    const float* gsrc,            // global address of tile start
    uint32_t     lds_byte,        // LDS byte offset of tile start
    uint32_t     tile_x,          // tile width  (elements, contiguous dim)
    uint32_t     tile_y,          // tile height (rows)
    uint32_t     tensor_x,        // tensor dim0 length (elements)
    uint32_t     tensor_y,        // tensor dim1 length (rows)
    uint32_t     row_stride,      // tensor dim0 stride (elements)
    bool         pad)             // +2 dwords every 32 dwords into LDS
{
    const uint64_t ga = (uint64_t)(uintptr_t)gsrc;

    u32x4 g0;
    g0.x = 1u;                                   // count=1, is_restore=0, no gather
    g0.y = lds_byte;                             // D#.lds_addr
    g0.z = (uint32_t)ga;                         // global_addr[31:0]
    g0.w = (uint32_t)(ga >> 32) | (2u << 30);    // global_addr[56:32] | type=2

    uint32_t w0 = (2u << 16);                    // data_size: 2 -> 4 bytes
    if (pad)                                     // pad_enable | interval(32dw) | amount(2dw)
        w0 |= (1u << 20) | (4u << 22) | (1u << 25);

    i32x8 g1;
    g1[0] = (int)w0;                                               // mask=0 | ds | pad
    g1[1] = (int)((tensor_x & 0xFFFFu) << 16);                     // barrier addr=0 | dim0 lo
    g1[2] = (int)((tensor_x >> 16) | ((tensor_y & 0xFFFFu) << 16));// dim0 hi | dim1 lo
    g1[3] = (int)((tensor_y >> 16) | (tile_x << 16));              // dim1 hi | tile_dim0
    g1[4] = (int)(tile_y & 0xFFFFu);                               // tile_dim1 | tile_dim2=0
    g1[5] = (int)row_stride;                                       // dim0_stride[31:0]
    g1[6] = 0;                                                     // stride hi | dim1_stride lo
    g1[7] = 0;                                                     // dim1_stride hi

    const i32x4 z4 = {0, 0, 0, 0};
#if defined(LRU_TDM_ARITY6)
    const i32x8 z8 = {0, 0, 0, 0, 0, 0, 0, 0};
    __builtin_amdgcn_tensor_load_to_lds(g0, g1, z4, z4, z8, 0);
#else
    __builtin_amdgcn_tensor_load_to_lds(g0, g1, z4, z4, 0);
#endif
}

// ---------------------------------------------------------------------------
// Stage 1:  x' = lambda * state + gamma * (inputs @ [B_re | B_im])
// ---------------------------------------------------------------------------
__global__ __launch_bounds__(256) void lru_stage1(
    const float* __restrict__ U,      // inputs [B, N]
    const float* __restrict__ Sre,    // state_re [B, H]
    const float* __restrict__ Sim,    // state_im [B, H]
    const float* __restrict__ Bre,    // [N, H] row-major
    const float* __restrict__ Bim,    // [N, H]
    const float* __restrict__ nu,     // [H]
    const float* __restrict__ theta,  // [H]
    const float* __restrict__ gamma,  // [H]
    float* __restrict__ Xre,          // out [B, H]
    float* __restrict__ Xim)          // out [B, H]
{
    // double-buffered: A-tile x2 (padded), B_re/B_im tiles x2
    __shared__ __align__(16) float smem[2 * S1_AS + 4 * S_BS];

    const int t    = threadIdx.x;
    const int w    = t >> 5;          // wave id 0..7
    const int lane = t & 31;
    const int half = lane >> 4;       // 0: K0/K1, 1: K2/K3
    const int l16  = lane & 15;
    const int m0   = blockIdx.y * BLOCK_ROWS;
    const int h0   = blockIdx.x * 16;

    auto issue = [&](int buf, int k0) {
        tdm_load_tile(U + (long)m0 * NREC + k0,
                      (uint32_t)(buf * S1_AS) * 4u,
                      TILE_K, BLOCK_ROWS, NREC, BATCH, NREC, /*pad=*/true);
        tdm_load_tile(Bre + (long)k0 * DHID + h0,
                      (uint32_t)(2 * S1_AS + buf * S_BS) * 4u,
                      16, TILE_K, DHID, NREC, DHID, /*pad=*/false);
        tdm_load_tile(Bim + (long)k0 * DHID + h0,
                      (uint32_t)(2 * S1_AS + 2 * S_BS + buf * S_BS) * 4u,
                      16, TILE_K, DHID, NREC, DHID, /*pad=*/false);
    };

    v8f accR = {};
    v8f accI = {};

    const int nch = NREC / TILE_K;
    if (w == 0) issue(0, 0);

    for (int ch = 0; ch < nch; ++ch) {
        const int cur = ch & 1;
        if (w == 0) {
            if (ch + 1 < nch) {
                issue(cur ^ 1, (ch + 1) * TILE_K);     // prefetch next chunk
                __builtin_amdgcn_s_wait_tensorcnt(3);  // current chunk's 3 loads done
            } else {
                __builtin_amdgcn_s_wait_tensorcnt(0);  // drain on last chunk
            }
        }
        __syncthreads();
        lds_publish(smem);   // TDM wrote smem behind the compiler's back

        const float* As  = smem + cur * S1_AS;
        const float* BsR = smem + 2 * S1_AS + cur * S_BS;
        const float* BsI = smem + 2 * S1_AS + 2 * S_BS + cur * S_BS;

        const int aRow = (w * 16 + l16) * LDS_AS;
        #pragma unroll
        for (int kk = 0; kk < TILE_K; kk += 4) {
            const int ka = kk + half * 2;
            v2f a;  a.x  = As[aRow + ka];        a.y  = As[aRow + ka + 1];
            v2f br; br.x = BsR[ka * 16 + l16];   br.y = BsR[(ka + 1) * 16 + l16];
            accR = wmma_f32(a, br, accR);
            v2f bi; bi.x = BsI[ka * 16 + l16];   bi.y = BsI[(ka + 1) * 16 + l16];
            accI = wmma_f32(a, bi, accI);
        }
        __syncthreads();
    }

    // ---- epilogue: diagonal complex recurrence, per-lane column h fixed ----
    const int   h  = h0 + l16;
    const float g  = gamma[h];
    const float lm = __expf(-__expf(nu[h]));
    const float lr = lm * __cosf(theta[h]);
    const float li = lm * __sinf(theta[h]);
    const int mBase = m0 + w * 16 + half * 8;
    #pragma unroll
    for (int r = 0; r < 8; ++r) {
        const long idx = (long)(mBase + r) * DHID + h;
        const float sr = Sre[idx];
        const float si = Sim[idx];
        Xre[idx] = sr * lr - si * li + g * accR[r];
        Xim[idx] = sr * li + si * lr + g * accI[r];
    }
}

// ---------------------------------------------------------------------------
// Stage 2:  y = x_re' @ C_re - x_im' @ C_im + D * inputs
// ---------------------------------------------------------------------------
__global__ __launch_bounds__(256) void lru_stage2(
    const float* __restrict__ Xre,   // [B, H]
    const float* __restrict__ Xim,   // [B, H]
    const float* __restrict__ Cre,   // [H, N] row-major
    const float* __restrict__ Cim,   // [H, N]
    const float* __restrict__ Dv,    // [N]
    const float* __restrict__ U,     // inputs [B, N]
    float* __restrict__ Y)           // out [B, N]
{
    // double-buffered: AsR x2, AsI x2 (padded), CsR x2, CsI x2
    __shared__ __align__(16) float smem[4 * S1_AS + 4 * S_BS];

    const int t    = threadIdx.x;
    const int w    = t >> 5;
    const int lane = t & 31;
    const int half = lane >> 4;
    const int l16  = lane & 15;
    const int m0   = blockIdx.y * BLOCK_ROWS;
    const int n0   = blockIdx.x * 16;

    auto issue = [&](int buf, int k0) {
        tdm_load_tile(Xre + (long)m0 * DHID + k0,
                      (uint32_t)(buf * S1_AS) * 4u,
                      TILE_K, BLOCK_ROWS, DHID, BATCH, DHID, /*pad=*/true);
        tdm_load_tile(Xim + (long)m0 * DHID + k0,
                      (uint32_t)(2 * S1_AS + buf * S1_AS) * 4u,
                      TILE_K, BLOCK_ROWS, DHID, BATCH, DHID, /*pad=*/true);
        tdm_load_tile(Cre + (long)k0 * NREC + n0,
                      (uint32_t)(4 * S1_AS + buf * S_BS) * 4u,
                      16, TILE_K, NREC, DHID, NREC, /*pad=*/false);
        tdm_load_tile(Cim + (long)k0 * NREC + n0,
                      (uint32_t)(4 * S1_AS + 2 * S_BS + buf * S_BS) * 4u,
                      16, TILE_K, NREC, DHID, NREC, /*pad=*/false);
    };

    v8f acc = {};

    const int nch = DHID / TILE_K;
    if (w == 0) issue(0, 0);

    for (int ch = 0; ch < nch; ++ch) {
        const int cur = ch & 1;
        if (w == 0) {
            if (ch + 1 < nch) {
                issue(cur ^ 1, (ch + 1) * TILE_K);     // prefetch next chunk
                __builtin_amdgcn_s_wait_tensorcnt(4);  // current chunk's 4 loads done
            } else {
                __builtin_amdgcn_s_wait_tensorcnt(0);  // drain on last chunk
            }
        }
        __syncthreads();
        lds_publish(smem);   // TDM wrote smem behind the compiler's back

        const float* AsR = smem + cur * S1_AS;
        const float* AsI = smem + 2 * S1_AS + cur * S1_AS;
        const float* CsR = smem + 4 * S1_AS + cur * S_BS;
        const float* CsI = smem + 4 * S1_AS + 2 * S_BS + cur * S_BS;

        const int aRow = (w * 16 + l16) * LDS_AS;
        #pragma unroll
        for (int kk = 0; kk < TILE_K; kk += 4) {
            const int ka = kk + half * 2;
            v2f ar; ar.x = AsR[aRow + ka];       ar.y = AsR[aRow + ka + 1];
            v2f br; br.x = CsR[ka * 16 + l16];   br.y = CsR[(ka + 1) * 16 + l16];
            acc = wmma_f32(ar, br, acc);
            v2f ai; ai.x = AsI[aRow + ka];       ai.y = AsI[aRow + ka + 1];
            v2f bi; bi.x = -CsI[ka * 16 + l16];  bi.y = -CsI[(ka + 1) * 16 + l16];
            acc = wmma_f32(ai, bi, acc);          // subtract x_im' @ C_im
        }
        __syncthreads();
    }

    // ---- epilogue: + D[n] * inputs[m][n] ----
    const int   n = n0 + l16;
    const float d = Dv[n];
    const int mBase = m0 + w * 16 + half * 8;
    #pragma unroll
    for (int r = 0; r < 8; ++r) {
        const long idx = (long)(mBase + r) * NREC + n;
        Y[idx] = acc[r] + d * U[idx];
    }
}

extern "C" void kernel_launch(void* const* d_in, const int* in_sizes, int n_in,
                              void* d_out, int out_size, void* d_ws, size_t ws_size,
                              hipStream_t stream) {
    (void)in_sizes; (void)n_in; (void)out_size; (void)d_ws; (void)ws_size;

    const float* inputs   = (const float*)d_in[0];
    const float* state_re = (const float*)d_in[1];
    const float* state_im = (const float*)d_in[2];
    const float* B_re     = (const float*)d_in[3];
    const float* B_im     = (const float*)d_in[4];
    const float* C_re     = (const float*)d_in[5];
    const float* C_im     = (const float*)d_in[6];
    const float* Dv       = (const float*)d_in[7];
    const float* nu       = (const float*)d_in[8];
    const float* theta    = (const float*)d_in[9];
    const float* gamma    = (const float*)d_in[10];

    float* y   = (float*)d_out;                       // [B, N]
    float* xre = y   + (size_t)BATCH * NREC;          // [B, H]
    float* xim = xre + (size_t)BATCH * DHID;          // [B, H]

    dim3 g1(DHID / 16, BATCH / BLOCK_ROWS);
    lru_stage1<<<g1, 256, 0, stream>>>(inputs, state_re, state_im,
                                       B_re, B_im, nu, theta, gamma, xre, xim);

    dim3 g2(NREC / 16, BATCH / BLOCK_ROWS);
    lru_stage2<<<g2, 256, 0, stream>>>(xre, xim, C_re, C_im, Dv, inputs, y);
}